// NRC_57956288692800
// MI455X (gfx1250) — compile-verified
//
#include <hip/hip_runtime.h>

// ---------------- constants ----------------
#define BATCH   256
#define DIMD    256          // feature dim (= GEMM K)
#define NCLS    64
#define NBANK   200000
#define KNEI    5
#define TOP     6            // K+1
#define NCHUNK  1024         // columns of bank handled per block
#define NCHUNKS ((NBANK + NCHUNK - 1) / NCHUNK)   // 196
#define NEG_INF (-3.402823466e38f)

typedef float v2f __attribute__((ext_vector_type(2)));
typedef float v8f __attribute__((ext_vector_type(8)));

// ---------------- small helpers ----------------
__device__ __forceinline__ void topk_insert(float (&rv)[TOP], int (&ri)[TOP],
                                            float v, int id) {
    if (v <= rv[TOP - 1]) return;          // strict >: equal keeps earlier (smaller) idx
    rv[TOP - 1] = v; ri[TOP - 1] = id;
#pragma unroll
    for (int j = TOP - 1; j > 0; --j) {
        if (rv[j] > rv[j - 1]) {
            float tv = rv[j]; rv[j] = rv[j - 1]; rv[j - 1] = tv;
            int   ti = ri[j]; ri[j] = ri[j - 1]; ri[j - 1] = ti;
        }
    }
}

// ---------------- K1: softmax + row-normalize ----------------
// grid = BATCH, block = 64 (thread == class)
__global__ void prep_kernel(const float* __restrict__ features,
                            const float* __restrict__ W,
                            const float* __restrict__ bias,
                            float* __restrict__ f_norm,
                            float* __restrict__ softmax_out) {
    const int b = blockIdx.x, t = threadIdx.x;
    __shared__ float feat[DIMD];
    __shared__ float red[64];

    float4 fv = *(const float4*)(features + b * DIMD + t * 4);
    *(float4*)(feat + t * 4) = fv;
    __syncthreads();

    float acc = bias[t];
#pragma unroll 8
    for (int d = 0; d < DIMD; ++d) acc += feat[d] * W[d * NCLS + t];

    red[t] = acc; __syncthreads();
    for (int s = 32; s > 0; s >>= 1) { if (t < s) red[t] = fmaxf(red[t], red[t + s]); __syncthreads(); }
    float mx = red[0]; __syncthreads();
    float e = expf(acc - mx);
    red[t] = e; __syncthreads();
    for (int s = 32; s > 0; s >>= 1) { if (t < s) red[t] += red[t + s]; __syncthreads(); }
    softmax_out[b * NCLS + t] = e / red[0];
    __syncthreads();

    red[t] = fv.x * fv.x + fv.y * fv.y + fv.z * fv.z + fv.w * fv.w;
    __syncthreads();
    for (int s = 32; s > 0; s >>= 1) { if (t < s) red[t] += red[t + s]; __syncthreads(); }
    float inv = 1.0f / fmaxf(sqrtf(red[0]), 1e-12f);
    float4 o; o.x = fv.x * inv; o.y = fv.y * inv; o.z = fv.z * inv; o.w = fv.w * inv;
    *(float4*)(f_norm + b * DIMD + t * 4) = o;
}

// ---------------- K2/K3: substitution map ----------------
__global__ void subst_init(int* __restrict__ subst) {
    int i = blockIdx.x * 256 + threadIdx.x;
    if (i < NBANK) subst[i] = -1;
}
// serial -> deterministic "last write wins" like jax .at[].set
__global__ void subst_scatter(const int* __restrict__ trg, int* __restrict__ subst) {
    if (blockIdx.x == 0 && threadIdx.x == 0)
        for (int b = 0; b < BATCH; ++b) subst[trg[b]] = b;
}

// ---------------- K4: fused WMMA GEMM (A @ bankT) + streaming top-6 ----------------
// grid = (NCHUNKS, M/128), block = 256 (8 waves)
// Each block: 128 rows x 1024 bank columns, processed in 16 sub-chunks of 64 cols.
// Bank rows with subst[n] >= 0 are redirected to f_norm (the fb substitution).
__global__ __launch_bounds__(256)
void gemm_topk_kernel(const float* __restrict__ A,        // M x 256 row-major
                      const float* __restrict__ fea_bank, // NBANK x 256
                      const float* __restrict__ f_norm,   // BATCH x 256
                      const int*   __restrict__ subst,
                      float* __restrict__ cand_val,       // M x (nchunks*6)
                      int*   __restrict__ cand_idx,
                      int nchunks) {
    __shared__ float smem[13056];   // A panel 128x68 (8704) + B panel 64x65 (4160)
    float* sA = smem;               // pitch 68
    float* sB = smem + 8704;        // pitch 65, stored transposed [k][n]
    float* sD = smem;               // 128x66 dist tile, reuses A region after GEMM
    __shared__ int sSub[64];

    const int chunk = blockIdx.x;
    const int mblk  = blockIdx.y;
    const int t     = threadIdx.x;
    const int lane  = t & 31;
    const int wave  = t >> 5;
    const int l16   = lane & 15;
    const int half  = lane >> 4;
    const float* Ablk = A + (size_t)(mblk * 128) * DIMD;

    float rv[TOP]; int ri[TOP];
#pragma unroll
    for (int j = 0; j < TOP; ++j) { rv[j] = NEG_INF; ri[j] = 0x7fffffff; }

    for (int sub = 0; sub < NCHUNK / 64; ++sub) {
        const int nbase = chunk * NCHUNK + sub * 64;
        if (t < 64) {
            int n = nbase + t;
            sSub[t] = (n < NBANK) ? subst[n] : -1;
        }
        __syncthreads();

        v8f acc0 = {}, acc1 = {}, acc2 = {}, acc3 = {};

        for (int kp = 0; kp < DIMD; kp += 64) {
            // ---- stage A panel: 128 rows x 64 cols (2048 float4, 8/thread) ----
#pragma unroll
            for (int i = 0; i < 8; ++i) {
                int idx = t + i * 256;
                int r = idx >> 4, c4 = (idx & 15) << 2;
                float4 v = *(const float4*)(Ablk + r * DIMD + kp + c4);
                *(float4*)(sA + r * 68 + c4) = v;
            }
            // ---- stage B panel transposed: 64 bank rows x 64 cols ----
#pragma unroll
            for (int i = 0; i < 4; ++i) {
                int idx = t + i * 256;
                int r = idx >> 4, c4 = (idx & 15) << 2;     // r = local bank row
                int n = nbase + r;
                float4 v = make_float4(0.f, 0.f, 0.f, 0.f);
                if (n < NBANK) {
                    int s = sSub[r];
                    const float* src = (s >= 0) ? (f_norm + (size_t)s * DIMD)
                                                : (fea_bank + (size_t)n * DIMD);
                    v = *(const float4*)(src + kp + c4);
                }
                sB[(c4 + 0) * 65 + r] = v.x;
                sB[(c4 + 1) * 65 + r] = v.y;
                sB[(c4 + 2) * 65 + r] = v.z;
                sB[(c4 + 3) * 65 + r] = v.w;
            }
            __syncthreads();

            // ---- WMMA: wave 'wave' -> rows [wave*16, wave*16+16), 4 n-tiles ----
            const float* aRow = sA + (wave * 16 + l16) * 68;
#pragma unroll
            for (int kk = 0; kk < 64; kk += 4) {
                v2f a; a.x = aRow[kk + 2 * half]; a.y = aRow[kk + 2 * half + 1];
                const float* b0r = sB + (kk + 2 * half) * 65;
                const float* b1r = b0r + 65;
                v2f b0; b0.x = b0r[l16];      b0.y = b1r[l16];
                v2f b1; b1.x = b0r[16 + l16]; b1.y = b1r[16 + l16];
                v2f b2; b2.x = b0r[32 + l16]; b2.y = b1r[32 + l16];
                v2f b3; b3.x = b0r[48 + l16]; b3.y = b1r[48 + l16];
                acc0 = __builtin_amdgcn_wmma_f32_16x16x4_f32(false, a, false, b0, (short)0, acc0, false, false);
                acc1 = __builtin_amdgcn_wmma_f32_16x16x4_f32(false, a, false, b1, (short)0, acc1, false, false);
                acc2 = __builtin_amdgcn_wmma_f32_16x16x4_f32(false, a, false, b2, (short)0, acc2, false, false);
                acc3 = __builtin_amdgcn_wmma_f32_16x16x4_f32(false, a, false, b3, (short)0, acc3, false, false);
            }
            __syncthreads();
        }

        // ---- spill 128x64 dist tile to LDS (C/D layout: vgpr j -> row half*8+j) ----
        float* dBase = sD + (wave * 16 + half * 8) * 66;
#pragma unroll
        for (int j = 0; j < 8; ++j) {
            float* dr = dBase + j * 66;
            dr[ 0 + l16] = acc0[j];
            dr[16 + l16] = acc1[j];
            dr[32 + l16] = acc2[j];
            dr[48 + l16] = acc3[j];
        }
        __syncthreads();

        // ---- streaming top-6 update: one thread per output row ----
        if (t < 128) {
            const float* dr = sD + t * 66;
            for (int c = 0; c < 64; ++c) {
                int n = nbase + c;
                if (n < NBANK) topk_insert(rv, ri, dr[c], n);
            }
        }
        __syncthreads();
    }

    if (t < 128) {
        int m = mblk * 128 + t;
        size_t base = ((size_t)m * nchunks + chunk) * TOP;
#pragma unroll
        for (int j = 0; j < TOP; ++j) { cand_val[base + j] = rv[j]; cand_idx[base + j] = ri[j]; }
    }
}

// ---------------- K5: merge per-chunk candidates, drop top-1 ----------------
// thread per row; out_idx is rows x 5
__global__ void merge_topk(const float* __restrict__ cand_val,
                           const int*   __restrict__ cand_idx,
                           int rows, int ncand,
                           int* __restrict__ out_idx) {
    int m = blockIdx.x * 64 + threadIdx.x;
    if (m >= rows) return;
    float rv[TOP]; int ri[TOP];
#pragma unroll
    for (int j = 0; j < TOP; ++j) { rv[j] = NEG_INF; ri[j] = 0x7fffffff; }
    const float* cv = cand_val + (size_t)m * ncand;
    const int*   ci = cand_idx + (size_t)m * ncand;
    for (int j = 0; j < ncand; ++j) topk_insert(rv, ri, cv[j], ci[j]);
#pragma unroll
    for (int j = 0; j < KNEI; ++j) out_idx[m * KNEI + j] = ri[j + 1];
}

// ---------------- K6: gather fea_near = fb[idx_near] ----------------
// grid = BATCH*KNEI, block = 64
__global__ void gather_fea(const int* __restrict__ idx_near,
                           const int* __restrict__ subst,
                           const float* __restrict__ fea_bank,
                           const float* __restrict__ f_norm,
                           float* __restrict__ fea_near) {
    int r = blockIdx.x;
    int i = idx_near[r];
    int s = subst[i];
    const float* src = (s >= 0) ? (f_norm + (size_t)s * DIMD)
                                : (fea_bank + (size_t)i * DIMD);
    int t = threadIdx.x;
    *(float4*)(fea_near + (size_t)r * DIMD + t * 4) = *(const float4*)(src + t * 4);
}

// ---------------- K7: per-batch KL terms ----------------
// grid = BATCH, block = 64 (thread == class)
__global__ void loss_b_kernel(const int* __restrict__ idx_near,   // BATCH x 5
                              const int* __restrict__ idx_nn,     // BATCH x 25
                              const int* __restrict__ subst,
                              const float* __restrict__ score_bank,
                              const float* __restrict__ softmax_out,
                              const int* __restrict__ trg,
                              float* __restrict__ loss_b) {
    const int b = blockIdx.x, c = threadIdx.x;
    __shared__ float red[64];
    const float p = softmax_out[b * NCLS + c];

    // term1: 0.1 * sum over 25 neighbors of sum_c s*(log s - p)
    float s1 = 0.f;
    for (int j = 0; j < 25; ++j) {
        int i = idx_nn[b * 25 + j];
        int s = subst[i];
        float v = (s >= 0) ? softmax_out[s * NCLS + c] : score_bank[(size_t)i * NCLS + c];
        s1 += v * (logf(v) - p);
    }
    red[c] = s1; __syncthreads();
    for (int s = 32; s > 0; s >>= 1) { if (c < s) red[c] += red[c + s]; __syncthreads(); }
    float t1 = 0.1f * red[0];
    __syncthreads();

    // term2: sum_k weight[b,k] * sum_c s*(log s - p)
    float t2 = 0.f;
    const int trgb = trg[b];
    for (int k = 0; k < KNEI; ++k) {
        int i = idx_near[b * KNEI + k];
        int s = subst[i];
        float v = (s >= 0) ? softmax_out[s * NCLS + c] : score_bank[(size_t)i * NCLS + c];
        red[c] = v * (logf(v) - p); __syncthreads();
        for (int st = 32; st > 0; st >>= 1) { if (c < st) red[c] += red[c + st]; __syncthreads(); }
        int match = 0;
#pragma unroll
        for (int k2 = 0; k2 < KNEI; ++k2) match += (idx_nn[b * 25 + k * KNEI + k2] == trgb);
        float w = (match > 0) ? (float)match : 0.1f;
        t2 += w * red[0];
        __syncthreads();
    }
    if (c == 0) loss_b[b] = t1 + t2;
}

// ---------------- K8: final reduce + gentropy ----------------
__global__ void final_kernel(const float* __restrict__ loss_b,
                             const float* __restrict__ softmax_out,
                             float* __restrict__ out) {
    __shared__ float red[256];
    const int t = threadIdx.x;
    red[t] = loss_b[t]; __syncthreads();
    for (int s = 128; s > 0; s >>= 1) { if (t < s) red[t] += red[t + s]; __syncthreads(); }
    float S = red[0];
    __syncthreads();
    float g = 0.f;
    if (t < NCLS) {
        float m = 0.f;
        for (int b = 0; b < BATCH; ++b) m += softmax_out[b * NCLS + t];
        m *= (1.0f / BATCH);
        g = m * logf(m + 1e-5f);
    }
    red[t] = g; __syncthreads();
    for (int s = 128; s > 0; s >>= 1) { if (t < s) red[t] += red[t + s]; __syncthreads(); }
    if (t == 0) out[0] = S * (1.0f / BATCH) + red[0];
}

// ---------------- host ----------------
static inline size_t align256(size_t x) { return (x + 255) & ~(size_t)255; }

extern "C" void kernel_launch(void* const* d_in, const int* in_sizes, int n_in,
                              void* d_out, int out_size, void* d_ws, size_t ws_size,
                              hipStream_t stream) {
    const float* features   = (const float*)d_in[0];
    const float* W          = (const float*)d_in[1];
    const float* bias       = (const float*)d_in[2];
    const float* fea_bank   = (const float*)d_in[3];
    const float* score_bank = (const float*)d_in[4];
    const int*   trg        = (const int*)d_in[5];
    float* out = (float*)d_out;
    (void)in_sizes; (void)n_in; (void)out_size; (void)ws_size;

    // workspace carve-out
    char* ws = (char*)d_ws;
    size_t off = 0;
    float* f_norm   = (float*)(ws + off); off = align256(off + (size_t)BATCH * DIMD * 4);
    float* softmax  = (float*)(ws + off); off = align256(off + (size_t)BATCH * NCLS * 4);
    int*   subst    = (int*)  (ws + off); off = align256(off + (size_t)NBANK * 4);
    const size_t candElems = (size_t)(BATCH * KNEI) * NCHUNKS * TOP;   // max rows = 1280
    float* cand_val = (float*)(ws + off); off = align256(off + candElems * 4);
    int*   cand_idx = (int*)  (ws + off); off = align256(off + candElems * 4);
    int*   idx_near = (int*)  (ws + off); off = align256(off + (size_t)BATCH * KNEI * 4);
    int*   idx_nn   = (int*)  (ws + off); off = align256(off + (size_t)BATCH * 25 * 4);
    float* fea_near = (float*)(ws + off); off = align256(off + (size_t)BATCH * KNEI * DIMD * 4);
    float* loss_b   = (float*)(ws + off); off = align256(off + (size_t)BATCH * 4);

    // 1) softmax + normalized features
    prep_kernel<<<BATCH, 64, 0, stream>>>(features, W, bias, f_norm, softmax);
    // 2) substitution map for fb/sb
    subst_init<<<(NBANK + 255) / 256, 256, 0, stream>>>(subst);
    subst_scatter<<<1, 64, 0, stream>>>(trg, subst);
    // 3) dist = f_norm @ fbT, fused streaming top-6
    gemm_topk_kernel<<<dim3(NCHUNKS, BATCH / 128), 256, 0, stream>>>(
        f_norm, fea_bank, f_norm, subst, cand_val, cand_idx, NCHUNKS);
    merge_topk<<<(BATCH + 63) / 64, 64, 0, stream>>>(
        cand_val, cand_idx, BATCH, NCHUNKS * TOP, idx_near);
    // 4) gather fea_near = fb[idx_near]
    gather_fea<<<BATCH * KNEI, 64, 0, stream>>>(idx_near, subst, fea_bank, f_norm, fea_near);
    // 5) dist2 = fea_near @ fbT, fused streaming top-6
    gemm_topk_kernel<<<dim3(NCHUNKS, (BATCH * KNEI) / 128), 256, 0, stream>>>(
        fea_near, fea_bank, f_norm, subst, cand_val, cand_idx, NCHUNKS);
    merge_topk<<<((BATCH * KNEI) + 63) / 64, 64, 0, stream>>>(
        cand_val, cand_idx, BATCH * KNEI, NCHUNKS * TOP, idx_nn);
    // 6) KL terms + gentropy -> scalar
    loss_b_kernel<<<BATCH, 64, 0, stream>>>(idx_near, idx_nn, subst, score_bank,
                                            softmax, trg, loss_b);
    final_kernel<<<1, 256, 0, stream>>>(loss_b, softmax, out);
}